// AnchorHead_54425825574969
// MI455X (gfx1250) — compile-verified
//
#include <hip/hip_runtime.h>
#include <hip/hip_bf16.h>

#define A_TOT   49104
#define B_IMG   16
#define M_GT    100
#define NCLS    80
#define C4      (NCLS / 4)                       // 20 float4 per anchor row
#define NVEC4   (B_IMG * A_TOT * C4)             // 15,713,280 float4 elements
#define ASSIGN_BX 192                            // ceil(49104/256)
#define FOCAL_BLOCKS 8192
#define BBOX_BLOCKS (ASSIGN_BX * B_IMG)          // 3072

typedef __attribute__((ext_vector_type(2))) float v2f;
typedef __attribute__((ext_vector_type(8))) float v8f;

// ---------------------------------------------------------------------------
// Block reduction (256 threads). First stage uses V_WMMA_F32_16X16X4_F32 with
// an all-ones A matrix: D[m][n] = sum_k B[k][n]. Each lane deposits its
// partial into one B slot (and 0 into the other), so the sum of the 16 column
// sums in D row 0 equals the 32-lane wave sum regardless of the exact B slot
// striping. D row 0 lives in VGPR0 of D for lanes 0..15.
// Requires EXEC all-ones: call from converged code, all threads participate.
// ---------------------------------------------------------------------------
__device__ __forceinline__ float block_reduce_256(float x) {
  __shared__ float red[8 * 16];
  __shared__ float outv;
  v2f a;  a[0] = 1.0f; a[1] = 1.0f;      // all-ones A (16x4 f32)
  v2f bm; bm[0] = x;   bm[1] = 0.0f;     // lane partial into one B slot
  v8f c = {};
  c = __builtin_amdgcn_wmma_f32_16x16x4_f32(false, a, false, bm,
                                            (short)0, c, false, false);
  int lane = threadIdx.x & 31;
  int wv   = threadIdx.x >> 5;
  if (lane < 16) red[wv * 16 + lane] = c[0];   // D[0][lane] (column sums)
  __syncthreads();
  if (threadIdx.x == 0) {
    float s = 0.0f;
#pragma unroll 4
    for (int i = 0; i < 128; ++i) s += red[i]; // fixed order -> deterministic
    outv = s;
  }
  __syncthreads();
  float r = outv;
  __syncthreads();                             // safe for back-to-back calls
  return r;
}

// ---------------------------------------------------------------------------
// RetinaNet grid anchor, computed analytically. Done in f64 then cast to f32
// to bit-match numpy's double-precision anchor generation in the reference
// (assignment thresholds 0.4/0.5 make anchors discretely sensitive).
// ---------------------------------------------------------------------------
__device__ __forceinline__ float4 anchor_of(int a) {
  int base, f, s;
  if      (a < 36864) { base = 0;     f = 64; s = 8;   }
  else if (a < 46080) { base = 36864; f = 32; s = 16;  }
  else if (a < 48384) { base = 46080; f = 16; s = 32;  }
  else if (a < 48960) { base = 48384; f = 8;  s = 64;  }
  else                { base = 48960; f = 4;  s = 128; }
  int al   = a - base;
  int cell = al / 9;
  int k    = al - cell * 9;
  int row  = cell / f;
  int col  = cell - row * f;
  int ri   = k / 3;           // ratio index (0.5, 1, 2)
  int si   = k - ri * 3;      // scale index
  const double SC[3] = {4.0, 5.0396841995794928, 6.3496042078727976}; // 4*2^(j/3)
  const double SR[3] = {0.7071067811865476, 1.0, 1.4142135623730951}; // sqrt(r)
  double cx = (col + 0.5) * (double)s;
  double cy = (row + 0.5) * (double)s;
  double sw = (double)s * SC[si];
  double w  = sw / SR[ri];
  double h  = sw * SR[ri];
  float4 r;
  r.x = (float)(cx - w * 0.5);
  r.y = (float)(cy - h * 0.5);
  r.z = (float)(cx + w * 0.5);
  r.w = (float)(cy + h * 0.5);
  return r;
}

// ---------------------------------------------------------------------------
// Kernel 0: zero the per-image positive counters.
// ---------------------------------------------------------------------------
__global__ void k_init(int* __restrict__ posCnt) {
  if (threadIdx.x < B_IMG) posCnt[threadIdx.x] = 0;
}

// ---------------------------------------------------------------------------
// Kernel 1: IoU argmax assignment. grid = (192, 16), block = 256.
// Writes matched index per (image, anchor): >=0 pos, -1 background, -2 ignore.
// Integer atomics for positive counts (deterministic).
// ---------------------------------------------------------------------------
__global__ void k_assign(const float* __restrict__ tboxes,
                         const unsigned char* __restrict__ gmask,
                         int* __restrict__ idxArr,
                         int* __restrict__ posCnt) {
  __shared__ float4 sg[M_GT];
  __shared__ float  sarea[M_GT];
  __shared__ int    smask[M_GT];
  __shared__ int    scnt;
  const int b = blockIdx.y;
  if (threadIdx.x == 0) scnt = 0;
  if (threadIdx.x < M_GT) {
    float4 g = ((const float4*)tboxes)[b * M_GT + threadIdx.x];
    sg[threadIdx.x]    = g;
    sarea[threadIdx.x] = (g.z - g.x) * (g.w - g.y);
    smask[threadIdx.x] = gmask[b * M_GT + threadIdx.x];
  }
  __syncthreads();

  const int a = blockIdx.x * blockDim.x + threadIdx.x;
  bool pos = false;
  if (a < A_TOT) {
    float4 an = anchor_of(a);
    float area_a = (an.z - an.x) * (an.w - an.y);
    float best_iou = -2.0f;
    int   best = 0;
    for (int m = 0; m < M_GT; ++m) {
      float4 g = sg[m];
      float ltx = fmaxf(an.x, g.x), lty = fmaxf(an.y, g.y);
      float rbx = fminf(an.z, g.z), rby = fminf(an.w, g.w);
      float w = fmaxf(rbx - ltx, 0.0f), h = fmaxf(rby - lty, 0.0f);
      float inter = w * h;
      float iou = inter / fmaxf(area_a + sarea[m] - inter, 1e-6f);
      if (!smask[m]) iou = -1.0f;
      if (iou > best_iou) { best_iou = iou; best = m; }   // first-max (argmax)
    }
    int outv;
    if      (best_iou >= 0.5f) { outv = best; pos = true; }
    else if (best_iou <  0.4f) { outv = -1; }
    else                       { outv = -2; }
    idxArr[(size_t)b * A_TOT + a] = outv;
  }
  if (pos) atomicAdd(&scnt, 1);
  __syncthreads();
  if (threadIdx.x == 0 && scnt) atomicAdd(&posCnt[b], scnt);
}

// ---------------------------------------------------------------------------
// Kernel 2: SmoothL1 bbox loss partials. grid = (192, 16), block = 256.
// ---------------------------------------------------------------------------
__device__ __forceinline__ float smooth_l1(float diff) {
  const float BETA = 1.0f / 9.0f;
  float d = fabsf(diff);
  return (d < BETA) ? ((0.5f * d) * d) / BETA : (d - 0.5f * BETA);
}

__global__ void k_bbox(const float* __restrict__ bbox_pred,
                       const float* __restrict__ tboxes,
                       const int* __restrict__ idxArr,
                       const int* __restrict__ posCnt,
                       float* __restrict__ partial) {
  const int b = blockIdx.y;
  const int a = blockIdx.x * blockDim.x + threadIdx.x;
  float val = 0.0f;
  if (a < A_TOT) {
    int idx = idxArr[(size_t)b * A_TOT + a];
    if (idx >= 0) {
      float4 an = anchor_of(a);
      float aw = an.z - an.x, ah = an.w - an.y;
      float ax = (an.x + an.z) * 0.5f, ay = (an.y + an.w) * 0.5f;
      float4 g = ((const float4*)tboxes)[b * M_GT + idx];
      float gw = fmaxf(g.z - g.x, 1e-6f), gh = fmaxf(g.w - g.y, 1e-6f);
      float gx = (g.x + g.z) * 0.5f,      gy = (g.y + g.w) * 0.5f;
      float t0 = (gx - ax) / aw;
      float t1 = (gy - ay) / ah;
      float t2 = logf(gw / aw);
      float t3 = logf(gh / ah);
      float4 p = ((const float4*)bbox_pred)[(size_t)b * A_TOT + a];
      val = smooth_l1(p.x - t0) + smooth_l1(p.y - t1) +
            smooth_l1(p.z - t2) + smooth_l1(p.w - t3);
      val /= fmaxf(1.0f, (float)posCnt[b]);
    }
  }
  float tot = block_reduce_256(val);
  if (threadIdx.x == 0) partial[blockIdx.y * gridDim.x + blockIdx.x] = tot;
}

// ---------------------------------------------------------------------------
// Kernel 3: sigmoid focal loss partials. Memory-bound 251 MB stream of
// cls_score: coalesced float4 grid-stride + global_prefetch one stride ahead.
// ---------------------------------------------------------------------------
__device__ __forceinline__ float focal_elem(float x, bool t) {
  float ax = fabsf(x);
  float L  = log1pf(expf(-ax));            // shared softplus tail
  float sp_px = fmaxf(x, 0.0f) + L;        // softplus(x)
  float sp_mx = fmaxf(-x, 0.0f) + L;       // softplus(-x)
  float p  = 1.0f / (1.0f + expf(-x));
  float bce = t ? sp_mx : sp_px;
  float pt  = t ? p : (1.0f - p);
  float at  = t ? 0.25f : 0.75f;
  float om  = 1.0f - pt;
  return at * om * om * bce;               // gamma = 2
}

__global__ void k_focal(const float* __restrict__ cls,
                        const int* __restrict__ labels,
                        const int* __restrict__ idxArr,
                        const int* __restrict__ posCnt,
                        float* __restrict__ partial) {
  const float4* c4p = (const float4*)cls;
  const int stride = gridDim.x * blockDim.x;
  float acc = 0.0f;
  for (int t = blockIdx.x * blockDim.x + threadIdx.x; t < NVEC4; t += stride) {
    if (t + stride < NVEC4) __builtin_prefetch(c4p + t + stride, 0, 3);
    float4 x = c4p[t];
    int alin = t / C4;
    int c0   = (t - alin * C4) * 4;
    int idx  = idxArr[alin];
    float w  = 0.0f;
    int label = -1;
    if (idx >= -1) {                       // matched + background contribute
      int b = alin / A_TOT;
      w = 1.0f / fmaxf(1.0f, (float)posCnt[b]);
      if (idx >= 0) label = labels[b * M_GT + idx];
    }
    float s = focal_elem(x.x, c0     == label) +
              focal_elem(x.y, c0 + 1 == label) +
              focal_elem(x.z, c0 + 2 == label) +
              focal_elem(x.w, c0 + 3 == label);
    acc += s * w;
  }
  float tot = block_reduce_256(acc);
  if (threadIdx.x == 0) partial[blockIdx.x] = tot;
}

// ---------------------------------------------------------------------------
// Kernel 4: final deterministic combine. 1 block, fixed-order sums.
// ---------------------------------------------------------------------------
__global__ void k_final(const float* __restrict__ pcls, int nc,
                        const float* __restrict__ pbb, int nb,
                        float* __restrict__ out) {
  float sc = 0.0f, sb = 0.0f;
  for (int i = threadIdx.x; i < nc; i += blockDim.x) sc += pcls[i];
  for (int i = threadIdx.x; i < nb; i += blockDim.x) sb += pbb[i];
  sc = block_reduce_256(sc);
  sb = block_reduce_256(sb);
  if (threadIdx.x == 0) {
    out[0] = sc * (1.0f / 16.0f);          // / batch
    out[1] = sb * (10.0f / 16.0f);         // * loss_weight / batch
  }
}

// ---------------------------------------------------------------------------
extern "C" void kernel_launch(void* const* d_in, const int* in_sizes, int n_in,
                              void* d_out, int out_size, void* d_ws, size_t ws_size,
                              hipStream_t stream) {
  const float*         cls  = (const float*)d_in[0];         // [16,49104,80]
  const float*         bbp  = (const float*)d_in[1];         // [16,49104,4]
  const float*         tbx  = (const float*)d_in[2];         // [16,100,4]
  const int*           tlb  = (const int*)d_in[3];           // [16,100]
  const unsigned char* msk  = (const unsigned char*)d_in[4]; // [16,100] bool
  float* out = (float*)d_out;

  char* ws = (char*)d_ws;
  int*   posCnt = (int*)ws;                                   // 16 ints
  int*   idxArr = (int*)(ws + 64);                            // B*A ints
  float* pcls   = (float*)(ws + 64 + (size_t)B_IMG * A_TOT * sizeof(int));
  float* pbb    = pcls + FOCAL_BLOCKS;

  k_init<<<1, 32, 0, stream>>>(posCnt);
  k_assign<<<dim3(ASSIGN_BX, B_IMG), 256, 0, stream>>>(tbx, msk, idxArr, posCnt);
  k_bbox<<<dim3(ASSIGN_BX, B_IMG), 256, 0, stream>>>(bbp, tbx, idxArr, posCnt, pbb);
  k_focal<<<FOCAL_BLOCKS, 256, 0, stream>>>(cls, tlb, idxArr, posCnt, pcls);
  k_final<<<1, 256, 0, stream>>>(pcls, FOCAL_BLOCKS, pbb, BBOX_BLOCKS, out);
}